// SelfAttention_63084479644162
// MI455X (gfx1250) — compile-verified
//
#include <hip/hip_runtime.h>
#include <math.h>

// Problem constants (match reference)
#define H_   4
#define S_   128
#define D_   32
#define B_   64
#define SD_  4096          // S*D
#define EMB_ 128           // H*D
#define HBSD (H_ * B_ * SD_)   // 1,048,576 floats per Q/K/V/xh buffer

typedef __attribute__((ext_vector_type(2))) float v2f;
typedef __attribute__((ext_vector_type(8))) float v8f;

// -------------------------------------------------------------------------
// Kernel A: rearrange x [B,S,H*D] -> xh [H][B][S*D] (contiguous per head)
// -------------------------------------------------------------------------
__global__ void __launch_bounds__(256) rearrange_kernel(const float* __restrict__ x,
                                                        float* __restrict__ xh) {
    int idx = blockIdx.x * blockDim.x + threadIdx.x;   // float4 index, [0, 262144)
    int i4  = idx * 4;
    int i   = i4 % SD_;          // feature index within head-row (= s*32 + d)
    int tmp = i4 / SD_;
    int b   = tmp % B_;
    int h   = tmp / B_;
    int s   = i / D_;
    int d   = i % D_;            // multiple of 4
    const float4* src = (const float4*)(x + (size_t)b * (S_ * EMB_) + s * EMB_ + h * D_ + d);
    float4*       dst = (float4*)(xh + (size_t)h * (B_ * SD_) + (size_t)b * SD_ + i);
    *dst = *src;
}

// -------------------------------------------------------------------------
// Kernel B: Q/K/V projections.  Y[h] = xh[h] (64x4096) @ W[h]^T (4096x4096) + b
// grid = (32, 12); blockIdx.y -> (head, which-of-QKV); 4 waves per block.
// Each wave: one 32-col N tile (2 B-fragments), all four 16-row M tiles.
//   - W streamed non-temporally from HBM, each byte read exactly once
//     (805MB total vs 192MB L2 -> NT hint keeps xh/QKV L2-resident).
//   - 6 loads per 8 WMMAs; prefetch ~1KB ahead on both weight row groups.
// -------------------------------------------------------------------------
__global__ void __launch_bounds__(128, 2) qkv_gemm_kernel(
    const float* __restrict__ xh,
    const float* __restrict__ WQ, const float* __restrict__ bQ,
    const float* __restrict__ WK, const float* __restrict__ bK,
    const float* __restrict__ WV, const float* __restrict__ bV,
    float* __restrict__ Qo, float* __restrict__ Ko, float* __restrict__ Vo)
{
    const int lane = threadIdx.x & 31;
    const int wave = threadIdx.x >> 5;   // 0..3
    const int y    = blockIdx.y;         // 0..11
    const int sel  = y % 3;              // 0=Q 1=K 2=V
    const int h    = y / 3;

    const float* W;  const float* bias;  float* out;
    if (sel == 0)      { W = WQ; bias = bQ; out = Qo; }
    else if (sel == 1) { W = WK; bias = bK; out = Ko; }
    else               { W = WV; bias = bV; out = Vo; }

    const float* Wh = W  + (size_t)h * SD_ * SD_;
    const float* Xh = xh + (size_t)h * (B_ * SD_);
    const int n0 = blockIdx.x * 128 + wave * 32;   // 32 output columns per wave

    const int lr   = lane & 15;           // N column (and M row within tile)
    const int koff = (lane >> 4) * 2;     // K sub-offset per lane half
    const int hi8  = (lane >> 4) * 8;     // M offset for C/D layout

    const float* wrow0 = Wh + (size_t)(n0 + lr) * SD_ + koff;        // cols n0..n0+15
    const float* wrow1 = wrow0 + (size_t)16 * SD_;                   // cols n0+16..n0+31
    const float* xrow  = Xh + (size_t)lr * SD_ + koff;               // batch rows lr (+16/32/48)

    v8f acc[8] = {v8f{}, v8f{}, v8f{}, v8f{}, v8f{}, v8f{}, v8f{}, v8f{}};

    for (int k0 = 0; k0 < SD_; k0 += 32) {
        // prefetch the weight stream ~1KB ahead (speculative; OOB is dropped)
        __builtin_prefetch(wrow0 + k0 + 256, 0, 0);
        __builtin_prefetch(wrow1 + k0 + 256, 0, 0);
        #pragma unroll
        for (int kk = 0; kk < 32; kk += 4) {
            const int k = k0 + kk;
            v2f b0 = __builtin_nontemporal_load((const v2f*)(wrow0 + k));
            v2f b1 = __builtin_nontemporal_load((const v2f*)(wrow1 + k));
            v2f a0 = *(const v2f*)(xrow + k);
            v2f a1 = *(const v2f*)(xrow + (size_t)16 * SD_ + k);
            v2f a2 = *(const v2f*)(xrow + (size_t)32 * SD_ + k);
            v2f a3 = *(const v2f*)(xrow + (size_t)48 * SD_ + k);
            acc[0] = __builtin_amdgcn_wmma_f32_16x16x4_f32(false, a0, false, b0, (short)0, acc[0], false, false);
            acc[1] = __builtin_amdgcn_wmma_f32_16x16x4_f32(false, a1, false, b0, (short)0, acc[1], false, false);
            acc[2] = __builtin_amdgcn_wmma_f32_16x16x4_f32(false, a2, false, b0, (short)0, acc[2], false, false);
            acc[3] = __builtin_amdgcn_wmma_f32_16x16x4_f32(false, a3, false, b0, (short)0, acc[3], false, false);
            acc[4] = __builtin_amdgcn_wmma_f32_16x16x4_f32(false, a0, false, b1, (short)0, acc[4], false, false);
            acc[5] = __builtin_amdgcn_wmma_f32_16x16x4_f32(false, a1, false, b1, (short)0, acc[5], false, false);
            acc[6] = __builtin_amdgcn_wmma_f32_16x16x4_f32(false, a2, false, b1, (short)0, acc[6], false, false);
            acc[7] = __builtin_amdgcn_wmma_f32_16x16x4_f32(false, a3, false, b1, (short)0, acc[7], false, false);
        }
    }

    // bias: each lane owns two output columns (n0+lr, n0+16+lr) across 8 rows
    const float bv0 = bias[(size_t)h * SD_ + n0 + lr];
    const float bv1 = bias[(size_t)h * SD_ + n0 + 16 + lr];

    float* outhb = out + (size_t)h * (B_ * SD_);   // [B][SD] for this head
    #pragma unroll
    for (int mt = 0; mt < 4; ++mt) {
        #pragma unroll
        for (int r = 0; r < 8; ++r) {
            int row = mt * 16 + r + hi8;           // batch index b
            outhb[(size_t)row * SD_ + n0 + lr]      = acc[mt][r]     + bv0;
            outhb[(size_t)row * SD_ + n0 + 16 + lr] = acc[mt + 4][r] + bv1;
        }
    }
}

// -------------------------------------------------------------------------
// Kernel C: per-(h,b) attention.  scores=Q K^T (WMMA, causal mask into LDS),
// softmax along the QUERY axis (reference axis=2), Z = P V (WMMA), *1/sqrt(D).
// One workgroup (8 waves) per (h,b); LDS scores 128 x (128+4) f32.
// -------------------------------------------------------------------------
#define PSTRIDE 132

__global__ void __launch_bounds__(256, 1) attention_kernel(
    const float* __restrict__ Qw, const float* __restrict__ Kw,
    const float* __restrict__ Vw, float* __restrict__ out)
{
    extern __shared__ float smem[];   // [128][PSTRIDE]

    const int tid  = threadIdx.x;
    const int lane = tid & 31;
    const int wave = tid >> 5;
    const int hb   = blockIdx.x;      // h*64 + b
    const int h    = hb >> 6;
    const int b    = hb & 63;

    const float* Q = Qw + (size_t)hb * SD_;   // [128][32]
    const float* K = Kw + (size_t)hb * SD_;
    const float* V = Vw + (size_t)hb * SD_;

    const int lr   = lane & 15;
    const int koff = (lane >> 4) * 2;
    const int hi8  = (lane >> 4) * 8;

    // ---- Phase 1: scores tile row-block m0 = wave*16, all 8 column tiles ----
    {
        const int m0 = wave * 16;
        const float* qrow = Q + (size_t)(m0 + lr) * D_ + koff;
        for (int j = 0; j < 8; ++j) {
            const int n0 = j * 16;
            const float* krow = K + (size_t)(n0 + lr) * D_ + koff;
            v8f acc = {};
            #pragma unroll
            for (int d0 = 0; d0 < D_; d0 += 4) {
                v2f a  = *(const v2f*)(qrow + d0);
                v2f bf = *(const v2f*)(krow + d0);
                acc = __builtin_amdgcn_wmma_f32_16x16x4_f32(false, a, false, bf, (short)0, acc, false, false);
            }
            const int kcol = n0 + lr;
            #pragma unroll
            for (int r = 0; r < 8; ++r) {
                const int q = m0 + r + hi8;
                smem[q * PSTRIDE + kcol] = (kcol > q) ? -__builtin_inff() : acc[r];
            }
        }
    }
    __syncthreads();

    // ---- Phase 2: softmax over q (per key column k); threads 0..127 ----
    if (tid < S_) {
        const int k = tid;
        float m = -__builtin_inff();
        for (int q = 0; q < S_; ++q) m = fmaxf(m, smem[q * PSTRIDE + k]);
        float sum = 0.f;
        for (int q = 0; q < S_; ++q) {
            float e = __expf(smem[q * PSTRIDE + k] - m);  // exp(-inf)=0 for masked
            smem[q * PSTRIDE + k] = e;
            sum += e;
        }
        const float inv = 1.0f / sum;
        for (int q = 0; q < S_; ++q) smem[q * PSTRIDE + k] *= inv;
    }
    __syncthreads();

    // ---- Phase 3: Z = P @ V  (M=128 q, N=32 d, K=128 k), scale, write out ----
    {
        const int m0 = wave * 16;
        #pragma unroll
        for (int nt = 0; nt < 2; ++nt) {
            const int d0 = nt * 16;
            v8f acc = {};
            #pragma unroll 4
            for (int k0 = 0; k0 < S_; k0 += 4) {
                v2f a = *(const v2f*)(&smem[(m0 + lr) * PSTRIDE + k0 + koff]);
                const float* vp = V + (size_t)(k0 + koff) * D_ + d0 + lr;
                v2f bf; bf.x = vp[0]; bf.y = vp[D_];
                acc = __builtin_amdgcn_wmma_f32_16x16x4_f32(false, a, false, bf, (short)0, acc, false, false);
            }
            #pragma unroll
            for (int r = 0; r < 8; ++r) {
                const int q = m0 + r + hi8;
                out[(size_t)b * (S_ * EMB_) + (size_t)q * EMB_ + h * D_ + d0 + lr] =
                    acc[r] * 0.17677669529663688f;   // 1/sqrt(32)
            }
        }
    }
}

// -------------------------------------------------------------------------
// Launch: inputs are (x, WQ, bQ, WK, bK, WV, bV), all f32.
// Workspace layout (floats): xh | Q | K | V  = 4 x 1,048,576 = 16 MB total.
// -------------------------------------------------------------------------
extern "C" void kernel_launch(void* const* d_in, const int* in_sizes, int n_in,
                              void* d_out, int out_size, void* d_ws, size_t ws_size,
                              hipStream_t stream) {
    const float* x  = (const float*)d_in[0];
    const float* WQ = (const float*)d_in[1];
    const float* bQ = (const float*)d_in[2];
    const float* WK = (const float*)d_in[3];
    const float* bK = (const float*)d_in[4];
    const float* WV = (const float*)d_in[5];
    const float* bV = (const float*)d_in[6];

    float* ws = (float*)d_ws;
    float* xh = ws;
    float* Qo = ws + (size_t)1 * HBSD;
    float* Ko = ws + (size_t)2 * HBSD;
    float* Vo = ws + (size_t)3 * HBSD;
    float* out = (float*)d_out;

    // A: rearrange (262144 float4 elements)
    rearrange_kernel<<<1024, 256, 0, stream>>>(x, xh);

    // B: QKV projections (HBM-bound weight stream, NT hints)
    dim3 gridB(32, 12);
    qkv_gemm_kernel<<<gridB, 128, 0, stream>>>(xh, WQ, bQ, WK, bK, WV, bV, Qo, Ko, Vo);

    // C: attention, one workgroup per (h,b)
    attention_kernel<<<H_ * B_, 256, S_ * PSTRIDE * sizeof(float), stream>>>(Qo, Ko, Vo, out);
}